// RandomRatingGenerator_66168266162303
// MI455X (gfx1250) — compile-verified
//
#include <hip/hip_runtime.h>
#include <stdint.h>

// ---------------------------------------------------------------------------
// Threefry-2x32 (exact JAX semantics). All key material is compile-time
// constant because the reference uses a fixed seed (42) and fixed shape.
// ---------------------------------------------------------------------------
struct U2 { uint32_t a, b; };

__host__ __device__ __forceinline__ constexpr uint32_t rotl32(uint32_t x, int r) {
  return (x << r) | (x >> (32 - r));
}

__host__ __device__ __forceinline__ constexpr U2
threefry2x32(uint32_t k0, uint32_t k1, uint32_t x0, uint32_t x1) {
  const uint32_t ks0 = k0, ks1 = k1, ks2 = k0 ^ k1 ^ 0x1BD11BDAu;
  x0 += ks0; x1 += ks1;
#define TF_ROUND(r) { x0 += x1; x1 = rotl32(x1, (r)); x1 ^= x0; }
  TF_ROUND(13) TF_ROUND(15) TF_ROUND(26) TF_ROUND(6)
  x0 += ks1; x1 += ks2 + 1u;
  TF_ROUND(17) TF_ROUND(29) TF_ROUND(16) TF_ROUND(24)
  x0 += ks2; x1 += ks0 + 2u;
  TF_ROUND(13) TF_ROUND(15) TF_ROUND(26) TF_ROUND(6)
  x0 += ks0; x1 += ks1 + 3u;
  TF_ROUND(17) TF_ROUND(29) TF_ROUND(16) TF_ROUND(24)
  x0 += ks1; x1 += ks2 + 4u;
  TF_ROUND(13) TF_ROUND(15) TF_ROUND(26) TF_ROUND(6)
  x0 += ks2; x1 += ks0 + 5u;
#undef TF_ROUND
  return U2{x0, x1};
}

// jax.random.key(42) -> key data (0, 42).
// split(key): threefry2x32((0,42), x0=[0,1], x1=[2,3]);
//   out = [F0(0,2), F0(1,3), F1(0,2), F1(1,3)] reshaped (2,2)
//   k1 = (F0(0,2), F0(1,3))  (higher bits), k2 = (F1(0,2), F1(1,3)) (lower).
constexpr U2 SPLIT_A = threefry2x32(0u, 42u, 0u, 2u);  // F(0,2)
constexpr U2 SPLIT_B = threefry2x32(0u, 42u, 1u, 3u);  // F(1,3)
constexpr uint32_t K1A = SPLIT_A.a, K1B = SPLIT_B.a;   // subkey 1
constexpr uint32_t K2A = SPLIT_A.b, K2B = SPLIT_B.b;   // subkey 2

constexpr uint32_t N_ROWS  = 1024u * 200u;   // 204800 tokens
constexpr uint32_t HALF    = N_ROWS / 2u;    // 102400 (counter split point)
constexpr uint32_t VOCAB   = 1000u;
constexpr uint32_t VEC4    = VOCAB / 4u;     // 250 float4 per row

// rating = 1 + ((hb % 5) + (lb % 5)) % 5   (2^32 mod 5 == 1 => multiplier 1)
__device__ __forceinline__ uint32_t rating_of(uint32_t i) {
  const bool lo  = i < HALF;
  const uint32_t x0 = lo ? i : i - HALF;
  const uint32_t x1 = lo ? i + HALF : i;
  const U2 h = threefry2x32(K1A, K1B, x0, x1);
  const U2 l = threefry2x32(K2A, K2B, x0, x1);
  const uint32_t hb = lo ? h.a : h.b;
  const uint32_t lb = lo ? l.a : l.b;
  return 1u + ((hb % 5u) + (lb % 5u)) % 5u;
}

typedef float v4f __attribute__((ext_vector_type(4)));

// One wave32 owns 32 rows: 32 threefry evals done in parallel (one per lane),
// then each row is streamed out as 250 coalesced b128 non-temporal stores.
__global__ __launch_bounds__(256) void
RandomRatingGenerator_onehot_kernel(float* __restrict__ out, uint32_t nRows) {
  const uint32_t lane     = threadIdx.x & 31u;
  const uint32_t waveGlob = blockIdx.x * (blockDim.x >> 5) + (threadIdx.x >> 5);
  const uint32_t rowBase  = waveGlob * 32u;
  if (rowBase >= nRows) return;

  const uint32_t myRow    = rowBase + lane;
  const uint32_t myRating = rating_of(myRow < nRows ? myRow : 0u);

  for (uint32_t k = 0; k < 32u; ++k) {
    const uint32_t row = rowBase + k;
    if (row >= nRows) break;
    const uint32_t rat = (uint32_t)__shfl((int)myRating, (int)k, 32);
    v4f* __restrict__ rowp = (v4f*)(out + (size_t)row * VOCAB);
#pragma unroll
    for (uint32_t j = 0; j < 8u; ++j) {          // 8*32 = 256 >= 250 slots
      const uint32_t c = lane + 32u * j;
      if (c < VEC4) {
        const uint32_t b4 = c << 2;
        v4f v;
        v.x = (b4 + 0u == rat) ? 1.0f : 0.0f;
        v.y = (b4 + 1u == rat) ? 1.0f : 0.0f;
        v.z = (b4 + 2u == rat) ? 1.0f : 0.0f;
        v.w = (b4 + 3u == rat) ? 1.0f : 0.0f;
        __builtin_nontemporal_store(v, rowp + c); // global_store_b128 th:NT
      }
    }
  }
}

extern "C" void kernel_launch(void* const* d_in, const int* in_sizes, int n_in,
                              void* d_out, int out_size, void* d_ws, size_t ws_size,
                              hipStream_t stream) {
  (void)d_in; (void)n_in; (void)d_ws; (void)ws_size; (void)out_size;
  // rows = B*S from the (unused-by-value) index input; falls back to constant.
  uint32_t nRows = (n_in > 0 && in_sizes && in_sizes[0] > 0)
                       ? (uint32_t)in_sizes[0] : N_ROWS;
  const uint32_t rowsPerBlock = 256u;  // 8 waves * 32 rows
  const uint32_t blocks = (nRows + rowsPerBlock - 1u) / rowsPerBlock;
  RandomRatingGenerator_onehot_kernel<<<blocks, 256, 0, stream>>>(
      (float*)d_out, nRows);
}